// Seq2seqAttentionDecoder_3169685864777
// MI455X (gfx1250) — compile-verified
//
#include <hip/hip_runtime.h>
#include <cstddef>
#include <cstdint>

typedef __attribute__((ext_vector_type(2))) float v2f;
typedef __attribute__((ext_vector_type(8))) float v8f;

#define Bb   32
#define TDEC 64
#define TENC 256
#define Hh   512
#define Ee   512
#define Vv   32000
#define Ll   2
#define NHh  8
#define DH   64

// ---------------------------------------------------------------------------
// Generic fp32 WMMA GEMM: C[M,N] = A[M,K] @ B[K,N] (+bias) using
// V_WMMA_F32_16X16X4_F32.
//  - Block = 128 threads = 4 waves. All waves share one 16-row A tile
//    (m0 = blockIdx.x*16); each wave owns a 64-wide N strip (grid.y = N/256).
//  - A is staged through LDS in 16-wide K chunks with the CDNA5 async path
//    (global_load_async_to_lds_b64, ASYNCcnt-tracked), DOUBLE-BUFFERED:
//    chunk c+1 is issued before waiting on chunk c (s_wait_asynccnt 1), so
//    the async copy latency overlaps the previous chunk's WMMA work.
//  - All 16 B fragments + both A ds-loads of a chunk are hoisted into
//    registers before the 16 WMMAs, so the loads clause up behind a single
//    wait instead of a load->wait->wmma lockstep.
//  - TRANSB: B stored [N,K] row-major (ldb=K) -> computes A @ B^T
//  - ACCUM : C += result
//  - PERM  : output row m remapped to (m%pB)*pT + (m/pB)  (fuses the
//            [T,B,V] -> [B,T,V] transpose into the logits store)
// ---------------------------------------------------------------------------
template <bool TRANSB, bool ACCUM, bool PERM>
__global__ __launch_bounds__(128) void wmma_gemm_f32(
    const float* __restrict__ A, const float* __restrict__ Bm,
    const float* __restrict__ bias, float* __restrict__ C,
    int K, int lda, int ldb, int ldc, int pT, int pB)
{
    __shared__ float As[2][16 * 16];

    const int tid  = threadIdx.x;        // 0..127
    const int lane = tid & 31;
    const int wave = tid >> 5;
    const int m0   = blockIdx.x * 16;
    const int n0   = (blockIdx.y * 4 + wave) * 64;
    const int lm   = lane & 15;
    const int koff = (lane >> 4) << 1;   // lanes 0-15: K=0,1 ; lanes 16-31: K=2,3
    const int hi8  = (lane >> 4) << 3;   // C rows: lanes 0-15 M=r, 16-31 M=r+8

    // async A-tile staging: thread -> (row, 2-float chunk)
    const int arow = tid >> 3;           // 0..15
    const int achk = (tid & 7) << 1;     // 0,2,...,14
    const float* Ag = A + (size_t)(m0 + arow) * lda + achk;
    const unsigned aoff0 = (unsigned)(size_t)(void*)&As[0][arow * 16 + achk];
    const unsigned aoff1 = (unsigned)(size_t)(void*)&As[1][arow * 16 + achk];

    v8f acc[4] = {v8f{}, v8f{}, v8f{}, v8f{}};

    const int nchunks = K >> 4;

    // prologue: issue chunk 0 into buffer 0
    asm volatile("global_load_async_to_lds_b64 %0, %1, off"
                 :: "v"(aoff0), "v"(Ag) : "memory");

    for (int c = 0; c < nchunks; ++c) {
        const int kb = c << 4;

        // issue chunk c+1 into the other buffer, then wait for chunk c only
        if (c + 1 < nchunks) {
            const unsigned noff = ((c + 1) & 1) ? aoff1 : aoff0;
            asm volatile("global_load_async_to_lds_b64 %0, %1, off"
                         :: "v"(noff), "v"(Ag + kb + 16) : "memory");
            asm volatile("s_wait_asynccnt 1" ::: "memory");
        } else {
            asm volatile("s_wait_asynccnt 0" ::: "memory");
        }
        __syncthreads();   // whole 16x16 A tile resident for all waves

        const float* Acur = (c & 1) ? &As[1][0] : &As[0][0];

        // ---- hoist A fragments (ds loads) ----
        v2f afr[4];
#pragma unroll
        for (int kk = 0; kk < 4; ++kk) {
            afr[kk].x = Acur[lm * 16 + kk * 4 + koff];
            afr[kk].y = Acur[lm * 16 + kk * 4 + koff + 1];
        }

        // ---- hoist all 16 B fragments for this chunk ----
        v2f bfr[4][4];
        if (TRANSB) {
#pragma unroll
            for (int j = 0; j < 4; ++j) {
                const float* Bp =
                    Bm + (size_t)(n0 + j * 16 + lm) * ldb + kb + koff;
#pragma unroll
                for (int kk = 0; kk < 4; ++kk) {
                    bfr[kk][j].x = Bp[kk * 4];
                    bfr[kk][j].y = Bp[kk * 4 + 1];
                }
            }
        } else {
#pragma unroll
            for (int kk = 0; kk < 4; ++kk) {
                const float* Bp =
                    Bm + (size_t)(kb + kk * 4 + koff) * ldb + n0 + lm;
#pragma unroll
                for (int j = 0; j < 4; ++j) {
                    bfr[kk][j].x = Bp[j * 16];
                    bfr[kk][j].y = Bp[j * 16 + ldb];
                }
            }
        }

        // ---- 16 back-to-back WMMAs ----
#pragma unroll
        for (int kk = 0; kk < 4; ++kk) {
#pragma unroll
            for (int j = 0; j < 4; ++j) {
                acc[j] = __builtin_amdgcn_wmma_f32_16x16x4_f32(
                    false, afr[kk], false, bfr[kk][j], (short)0, acc[j],
                    false, false);
            }
        }
        __syncthreads();   // all waves done reading before buffer reuse
    }

#pragma unroll
    for (int j = 0; j < 4; ++j) {
        const int n = n0 + j * 16 + lm;
        const float bv = bias ? bias[n] : 0.0f;
#pragma unroll
        for (int r = 0; r < 8; ++r) {
            const int m = m0 + hi8 + r;
            const int row = PERM ? ((m % pB) * pT + (m / pB)) : m;
            const size_t off = (size_t)row * ldc + n;
            const float v = acc[j][r] + bv;
            if (ACCUM) C[off] += v;
            else       C[off]  = v;
        }
    }
}

// ---------------------------------------------------------------------------
// emb_seq[t][b][:] = emb[x[b][t]][:]   (T,B,E)
// ---------------------------------------------------------------------------
__global__ __launch_bounds__(256) void embed_k(
    const int* __restrict__ x, const float* __restrict__ emb,
    float* __restrict__ out)
{
    const int idx = blockIdx.x * 256 + threadIdx.x;   // < T*B*E
    const int e  = idx & (Ee - 1);
    const int tb = idx >> 9;                          // t*32 + b
    const int b  = tb & (Bb - 1);
    const int t  = tb >> 5;
    const int tok = x[b * TDEC + t];
    out[idx] = emb[(size_t)tok * Ee + e];
}

// ---------------------------------------------------------------------------
// Single-query masked multi-head attention, one (b, head) per block.
// ---------------------------------------------------------------------------
__global__ __launch_bounds__(256) void attn_step_k(
    const float* __restrict__ q, const float* __restrict__ Kh,
    const float* __restrict__ Vh, const int* __restrict__ valid,
    float* __restrict__ ctx)
{
    __shared__ float sc[TENC];
    __shared__ float red[256];
    const int b   = blockIdx.x;
    const int hd  = blockIdx.y;
    const int tid = threadIdx.x;
    const float scale = 0.125f;  // 1/sqrt(64)

    const float* qv = q + (size_t)b * Hh + hd * DH;
    const float* kv = Kh + ((size_t)(b * TENC + tid)) * Hh + hd * DH;
    float s = 0.0f;
#pragma unroll 8
    for (int d = 0; d < DH; ++d) s += qv[d] * kv[d];
    s *= scale;
    if (tid >= valid[b]) s = -1e6f;
    sc[tid]  = s;
    red[tid] = s;
    __syncthreads();
    for (int o = 128; o > 0; o >>= 1) {
        if (tid < o) red[tid] = fmaxf(red[tid], red[tid + o]);
        __syncthreads();
    }
    const float mx = red[0];
    __syncthreads();
    const float e = __expf(sc[tid] - mx);
    sc[tid]  = e;
    red[tid] = e;
    __syncthreads();
    for (int o = 128; o > 0; o >>= 1) {
        if (tid < o) red[tid] += red[tid + o];
        __syncthreads();
    }
    const float inv = 1.0f / red[0];
    __syncthreads();

    if (tid < DH) {
        const float* vb = Vh + ((size_t)b * TENC) * Hh + hd * DH + tid;
        float acc = 0.0f;
        for (int s2 = 0; s2 < TENC; ++s2) acc += sc[s2] * vb[(size_t)s2 * Hh];
        ctx[(size_t)b * Hh + hd * DH + tid] = acc * inv;
    }
}

// ---------------------------------------------------------------------------
// Per-row LayerNorm over H=512, one block per row (256 threads x 2 elems)
// ---------------------------------------------------------------------------
__global__ __launch_bounds__(256) void layernorm_k(
    const float* __restrict__ x, const float* __restrict__ g,
    const float* __restrict__ bt, float* __restrict__ y)
{
    __shared__ float r1[256];
    __shared__ float r2[256];
    const int b   = blockIdx.x;
    const int tid = threadIdx.x;
    const float* xr = x + (size_t)b * Hh;
    const float v0 = xr[tid];
    const float v1 = xr[tid + 256];
    r1[tid] = v0 + v1;
    r2[tid] = v0 * v0 + v1 * v1;
    __syncthreads();
    for (int o = 128; o > 0; o >>= 1) {
        if (tid < o) { r1[tid] += r1[tid + o]; r2[tid] += r2[tid + o]; }
        __syncthreads();
    }
    const float mean = r1[0] * (1.0f / Hh);
    const float var  = r2[0] * (1.0f / Hh) - mean * mean;
    const float inv  = rsqrtf(var + 1e-5f);
    y[(size_t)b * Hh + tid]       = (v0 - mean) * inv * g[tid] + bt[tid];
    y[(size_t)b * Hh + tid + 256] = (v1 - mean) * inv * g[tid + 256] + bt[tid + 256];
}

// ---------------------------------------------------------------------------
// GRU gate pointwise math + alpha-residual into xr; optional ys store.
// ---------------------------------------------------------------------------
__global__ __launch_bounds__(256) void gru_point_k(
    const float* __restrict__ gi, const float* __restrict__ gh,
    const float* __restrict__ bih, const float* __restrict__ bhh,
    float* __restrict__ h, float* __restrict__ xr,
    const float* __restrict__ alphas, int layer, float* __restrict__ ys)
{
    const int idx = blockIdx.x * 256 + threadIdx.x;   // < B*H
    const int b = idx >> 9;
    const int j = idx & (Hh - 1);
    const float* gib = gi + (size_t)b * (3 * Hh);
    const float* ghb = gh + (size_t)b * (3 * Hh);
    const float ir = gib[j]          + bih[j];
    const float iz = gib[Hh + j]     + bih[Hh + j];
    const float ig = gib[2 * Hh + j] + bih[2 * Hh + j];
    const float hr = ghb[j]          + bhh[j];
    const float hz = ghb[Hh + j]     + bhh[Hh + j];
    const float hg = ghb[2 * Hh + j] + bhh[2 * Hh + j];
    const float r = 1.0f / (1.0f + __expf(-(ir + hr)));
    const float z = 1.0f / (1.0f + __expf(-(iz + hz)));
    const float g = tanhf(ig + r * hg);
    const float hv = h[idx];
    const float hn = (1.0f - z) * g + z * hv;
    h[idx] = hn;
    const float nx = xr[idx] + alphas[layer] * hn;
    xr[idx] = nx;
    if (ys) ys[idx] = nx;
}

// ---------------------------------------------------------------------------
extern "C" void kernel_launch(void* const* d_in, const int* in_sizes, int n_in,
                              void* d_out, int out_size, void* d_ws, size_t ws_size,
                              hipStream_t stream) {
    const int*   x      = (const int*)  d_in[0];
    const float* enc    = (const float*)d_in[1];
    const float* h0     = (const float*)d_in[2];
    const int*   vlen   = (const int*)  d_in[3];
    const float* emb    = (const float*)d_in[4];
    const float* W_q    = (const float*)d_in[5];
    const float* W_k    = (const float*)d_in[6];
    const float* W_v    = (const float*)d_in[7];
    const float* W_o    = (const float*)d_in[8];
    const float* W_in   = (const float*)d_in[9];
    const float* ln_g   = (const float*)d_in[10];
    const float* ln_b   = (const float*)d_in[11];
    const float* alphas = (const float*)d_in[12];
    const float* W_ih   = (const float*)d_in[13];
    const float* W_hh   = (const float*)d_in[14];
    const float* b_ih   = (const float*)d_in[15];
    const float* b_hh   = (const float*)d_in[16];
    const float* W_d    = (const float*)d_in[17];
    const float* b_d    = (const float*)d_in[18];
    float* out = (float*)d_out;

    // -------- workspace carve-up (floats) --------
    float* ws = (float*)d_ws;
    const size_t SZ_KH  = (size_t)Bb * TENC * Hh;
    const size_t SZ_EMB = (size_t)TDEC * Bb * Ee;
    const size_t SZ_YS  = (size_t)TDEC * Bb * Hh;
    const size_t SZ_H   = (size_t)Ll * Bb * Hh;
    const size_t SZ_BH  = (size_t)Bb * Hh;
    const size_t SZ_G   = (size_t)Bb * 3 * Hh;
    float* Kh     = ws;               ws += SZ_KH;
    float* Vh     = ws;               ws += SZ_KH;
    float* embseq = ws;               ws += SZ_EMB;
    float* ys     = ws;               ws += SZ_YS;
    float* hbuf   = ws;               ws += SZ_H;
    float* qbuf   = ws;               ws += SZ_BH;
    float* ctx    = ws;               ws += SZ_BH;
    float* ao     = ws;               ws += SZ_BH;
    float* xr     = ws;               ws += SZ_BH;
    float* xin    = ws;               ws += SZ_BH;
    float* gi     = ws;               ws += SZ_G;
    float* gh     = ws;               ws += SZ_G;

    auto gemm = [&](const float* A, const float* Bm, const float* bias, float* C,
                    int M, int N, int K, int lda, int ldb, int ldc,
                    bool transB, bool accum, bool perm, int pT, int pB) {
        dim3 grid(M / 16, N / 256);
        if (transB)
            wmma_gemm_f32<true, false, false><<<grid, 128, 0, stream>>>(
                A, Bm, bias, C, K, lda, ldb, ldc, pT, pB);
        else if (accum)
            wmma_gemm_f32<false, true, false><<<grid, 128, 0, stream>>>(
                A, Bm, bias, C, K, lda, ldb, ldc, pT, pB);
        else if (perm)
            wmma_gemm_f32<false, false, true><<<grid, 128, 0, stream>>>(
                A, Bm, bias, C, K, lda, ldb, ldc, pT, pB);
        else
            wmma_gemm_f32<false, false, false><<<grid, 128, 0, stream>>>(
                A, Bm, bias, C, K, lda, ldb, ldc, pT, pB);
    };

    // -------- init: hidden state + embedding gather --------
    hipMemcpyAsync(hbuf, h0, SZ_H * sizeof(float), hipMemcpyDeviceToDevice, stream);
    embed_k<<<(TDEC * Bb * Ee) / 256, 256, 0, stream>>>(x, emb, embseq);

    // -------- hoisted K/V projections: (B*T_ENC,2H) @ (2H,H) --------
    gemm(enc, W_k, nullptr, Kh, Bb * TENC, Hh, 2 * Hh, 2 * Hh, Hh, Hh,
         false, false, false, 0, 0);
    gemm(enc, W_v, nullptr, Vh, Bb * TENC, Hh, 2 * Hh, 2 * Hh, Hh, Hh,
         false, false, false, 0, 0);

    // -------- sequential decode over T_DEC steps --------
    for (int t = 0; t < TDEC; ++t) {
        const float* e_t = embseq + (size_t)t * Bb * Ee;
        float* ys_t = ys + (size_t)t * Bb * Hh;

        // q = h[L-1] @ W_q
        gemm(hbuf + (size_t)(Ll - 1) * Bb * Hh, W_q, nullptr, qbuf,
             Bb, Hh, Hh, Hh, Hh, Hh, false, false, false, 0, 0);
        // masked MHA against encoder K/V
        attn_step_k<<<dim3(Bb, NHh), 256, 0, stream>>>(qbuf, Kh, Vh, vlen, ctx);
        // ao = ctx @ W_o
        gemm(ctx, W_o, nullptr, ao, Bb, Hh, Hh, Hh, Hh, Hh,
             false, false, false, 0, 0);
        // xr = [ao, e_t] @ W_in  (two GEMMs; second accumulates)
        gemm(ao, W_in, nullptr, xr, Bb, Hh, Hh, Hh, Hh, Hh,
             false, false, false, 0, 0);
        gemm(e_t, W_in + (size_t)Hh * Hh, nullptr, xr, Bb, Hh, Hh, Hh, Hh, Hh,
             false, true, false, 0, 0);

        for (int l = 0; l < Ll; ++l) {
            layernorm_k<<<Bb, 256, 0, stream>>>(xr, ln_g + (size_t)l * Hh,
                                                ln_b + (size_t)l * Hh, xin);
            // gi = xin @ W_ih[l]^T   (W_ih[l]: [3H,H] row-major -> transB)
            gemm(xin, W_ih + (size_t)l * 3 * Hh * Hh, nullptr, gi,
                 Bb, 3 * Hh, Hh, Hh, Hh, 3 * Hh, true, false, false, 0, 0);
            // gh = h[l] @ W_hh[l]^T
            gemm(hbuf + (size_t)l * Bb * Hh, W_hh + (size_t)l * 3 * Hh * Hh,
                 nullptr, gh, Bb, 3 * Hh, Hh, Hh, Hh, 3 * Hh,
                 true, false, false, 0, 0);
            gru_point_k<<<(Bb * Hh) / 256, 256, 0, stream>>>(
                gi, gh, b_ih + (size_t)l * 3 * Hh, b_hh + (size_t)l * 3 * Hh,
                hbuf + (size_t)l * Bb * Hh, xr, alphas, l,
                (l == Ll - 1) ? ys_t : nullptr);
        }
    }

    // -------- logits = ys @ W_dense + b, fused [T,B,V]->[B,T,V] store --------
    gemm(ys, W_d, b_d, out, TDEC * Bb, Vv, Hh, Hh, Vv, Vv,
         false, false, true, TDEC, Bb);

    // -------- hN tail --------
    hipMemcpyAsync(out + (size_t)Bb * TDEC * Vv, hbuf, SZ_H * sizeof(float),
                   hipMemcpyDeviceToDevice, stream);
}